// ORGaNICs2Dspectra_5248450035872
// MI455X (gfx1250) — compile-verified
//
#include <hip/hip_runtime.h>
#include <hip/hip_bf16.h>

typedef __attribute__((ext_vector_type(2))) float v2f;
typedef __attribute__((ext_vector_type(8))) float v8f;

#define NH 128          // n
#define N2 256          // 2n
#define BB 16           // batch
#define MM 32           // frequencies
#define LD 256
#define MAT_ELEMS (N2 * N2)         // 65536
#define SLOT_FLOATS (2 * MAT_ELEMS) // re plane + im plane
#define SMAX 240                    // max trailing extent
#define SSTR 18                     // padded LDS row stride (bank-conflict free, 8B aligned pairs)

// ---------------------------------------------------------------------------
// Kernel 1: steady state.  One block per batch, 128 threads (thread i = unit i)
// ---------------------------------------------------------------------------
__global__ void organics_precompute(const float* __restrict__ x,
                                    const float* __restrict__ Wzx,
                                    const float* __restrict__ logW,
                                    const float* __restrict__ b0,
                                    const float* __restrict__ sigma,
                                    float* __restrict__ ya) {
  const int b = blockIdx.x;
  const int i = threadIdx.x;  // 0..127
  __shared__ float g[NH];
  const float* xb = x + b * 256;
  const float* wr = Wzx + i * 256;
  float z = 0.0f;
  for (int k = 0; k < 256; ++k) z = fmaf(xb[k], wr[k], z);
  float B0 = 1.0f / (1.0f + __expf(-b0[i]));
  float rz = fmaxf(z, 0.0f);
  float gated = B0 * B0 * rz * rz;
  g[i] = gated;
  __syncthreads();
  float pooled = 0.0f;
  const float* lw = logW + i * NH;
  for (int j = 0; j < NH; ++j) pooled = fmaf(__expf(lw[j]), g[j], pooled);
  float sb = sigma[0] * B0;
  float denom = sb * sb + pooled;
  ya[b * N2 + i]      = gated / denom;  // y
  ya[b * N2 + NH + i] = denom;          // a (>0 always)
}

// ---------------------------------------------------------------------------
// Kernel 2: assemble Jacobian rows directly into d_out (b, 256, 256) row-major.
// ---------------------------------------------------------------------------
__global__ void organics_build_jac(const float* __restrict__ ya,
                                   const float* __restrict__ logW,
                                   const float* __restrict__ log_tauy,
                                   const float* __restrict__ log_taua,
                                   float* __restrict__ jac) {
  const int row = blockIdx.x;
  const int b   = blockIdx.y;
  const int col = threadIdx.x;
  const float inv_tauy = __expf(-log_tauy[0]);
  const float inv_taua = __expf(-log_taua[0]);
  const float* y = ya + b * N2;
  const float* a = y + NH;
  float val = 0.0f;
  if (row < NH) {
    float sa = sqrtf(fmaxf(a[row], 0.0f));
    if (col == row)           val = -sa * inv_tauy;
    else if (col == NH + row) val = -y[row] * 0.5f / sa * inv_tauy;
  } else {
    int p = row - NH;
    if (col < NH) {
      float w  = __expf(logW[p * NH + col]);
      float ry = fmaxf(y[col], 0.0f);
      val = w * 2.0f * a[col] * ry * inv_taua;
    } else {
      int j = col - NH;
      float w  = __expf(logW[p * NH + j]);
      float ry = fmaxf(y[j], 0.0f);
      val = (w * ry * ry - (p == j ? 1.0f : 0.0f)) * inv_taua;
    }
  }
  jac[(size_t)b * MAT_ELEMS + row * N2 + col] = val;
}

// ---------------------------------------------------------------------------
// Kernel 3: per (batch, omega) solve (J^T + i w I) v = e by blocked complex LU.
// Panel width 16; trailing update = complex GEMM on V_WMMA_F32_16X16X4_F32 with
// LDS-staged operands (L21 and U12^T, padded stride); S = sum(eta^2|v|^2)/n^2.
// grid = (8 freq-groups, 16 batches), block = 256 threads (8 waves).
// ---------------------------------------------------------------------------
__global__ void __launch_bounds__(256)
organics_solve(const float* __restrict__ jac,
               const float* __restrict__ omega,
               const float* __restrict__ eta,
               float* __restrict__ mats,
               float* __restrict__ S) {
  const int b    = blockIdx.y;
  const int slot = blockIdx.y * gridDim.x + blockIdx.x;
  float* __restrict__ Mre = mats + (size_t)slot * SLOT_FLOATS;
  float* __restrict__ Mim = Mre + MAT_ELEMS;
  const int tid  = threadIdx.x;
  const int lane = tid & 31;
  const int wv   = tid >> 5;         // wave id 0..7
  const int mrow = lane & 15;        // M (for A) / N (for B,C,D) index
  const int hsel = lane >> 4;        // half-wave select (K-pair / M-half)

  __shared__ float L11r[16][16], L11i[16][16];
  __shared__ float Ls_r[SMAX * SSTR], Ls_i[SMAX * SSTR];  // L21, [r][k]
  __shared__ float Ut_r[SMAX * SSTR], Ut_i[SMAX * SSTR];  // U12 transposed, [c][k]
  __shared__ float zr_s[N2], zi_s[N2];
  __shared__ float red[256];

  const float* jb = jac + (size_t)b * MAT_ELEMS;

  for (int mloc = 0; mloc < 4; ++mloc) {
    const int   m  = blockIdx.x * 4 + mloc;
    const float om = omega[m];

    // ---- init C = J^T + i*om*I ----
    for (int idx = tid; idx < MAT_ELEMS; idx += 256) {
      int r = idx >> 8, c = idx & 255;
      Mre[idx] = jb[c * N2 + r];            // transpose
      Mim[idx] = (r == c) ? om : 0.0f;
    }
    __threadfence_block();
    __syncthreads();

    // ---- blocked LU, no pivoting ----
    for (int k0 = 0; k0 < N2; k0 += 16) {
      // (a) panel factorization, columns k0..k0+15, rows below diagonal
      for (int j = k0; j < k0 + 16; ++j) {
        __syncthreads();
        float pr = Mre[j * LD + j], pi = Mim[j * LD + j];
        float inv = 1.0f / (pr * pr + pi * pi);
        float rr = pr * inv, ri = -pi * inv;           // 1/pivot
        for (int i = j + 1 + tid; i < N2; i += 256) {
          float ar = Mre[i * LD + j], ai = Mim[i * LD + j];
          float lr = ar * rr - ai * ri;
          float li = ar * ri + ai * rr;
          Mre[i * LD + j] = lr;
          Mim[i * LD + j] = li;
          for (int jj = j + 1; jj < k0 + 16; ++jj) {
            float ur = Mre[j * LD + jj], ui = Mim[j * LD + jj];
            Mre[i * LD + jj] -= lr * ur - li * ui;
            Mim[i * LD + jj] -= lr * ui + li * ur;
          }
        }
        __threadfence_block();
      }
      __syncthreads();

      // (b) stage unit-lower L11 (16x16) into LDS
      {
        int i = tid >> 4, j = tid & 15;
        L11r[i][j] = Mre[(k0 + i) * LD + (k0 + j)];
        L11i[i][j] = Mim[(k0 + i) * LD + (k0 + j)];
      }
      __syncthreads();

      // (c) U12 = L11^-1 * C12 (per-column local triangular solve)
      for (int c = k0 + 16 + tid; c < N2; c += 256) {
        float xr[16], xi[16];
        for (int i = 0; i < 16; ++i) {
          xr[i] = Mre[(k0 + i) * LD + c];
          xi[i] = Mim[(k0 + i) * LD + c];
        }
        for (int i = 1; i < 16; ++i)
          for (int j = 0; j < i; ++j) {
            xr[i] -= L11r[i][j] * xr[j] - L11i[i][j] * xi[j];
            xi[i] -= L11r[i][j] * xi[j] + L11i[i][j] * xr[j];
          }
        for (int i = 0; i < 16; ++i) {
          Mre[(k0 + i) * LD + c] = xr[i];
          Mim[(k0 + i) * LD + c] = xi[i];
        }
      }
      __threadfence_block();
      __syncthreads();

      const int s   = N2 - k0 - 16;
      const int t16 = s >> 4;

      // (d) stage L21 (s x 16) and U12^T (s x 16) into padded LDS tiles:
      // one global read each; all WMMA operand traffic then comes from LDS.
      for (int idx = tid; idx < s * 16; idx += 256) {
        int r = idx >> 4, k = idx & 15;
        Ls_r[r * SSTR + k] = Mre[(k0 + 16 + r) * LD + (k0 + k)];
        Ls_i[r * SSTR + k] = Mim[(k0 + 16 + r) * LD + (k0 + k)];
        Ut_r[r * SSTR + k] = Mre[(k0 + k) * LD + (k0 + 16 + r)];
        Ut_i[r * SSTR + k] = Mim[(k0 + k) * LD + (k0 + 16 + r)];
      }
      __syncthreads();

      // (e) trailing update C22 -= L21 * U12 via complex GEMM on f32 WMMA.
      // Each wave owns 16x16 output tiles; K = 16 done as 4x K=4 WMMA steps.
      for (int t = wv; t < t16 * t16; t += 8) {     // wave-uniform loop
        int ti = t / t16, tj = t - ti * t16;
        int r0 = k0 + 16 + ti * 16;
        int c0 = k0 + 16 + tj * 16;
        int la = (ti * 16 + mrow) * SSTR;           // A row in staged L21
        int lb = (tj * 16 + mrow) * SSTR;           // B col in staged U12^T
        v8f arr = {}, aii = {}, ari = {}, air = {};
#pragma unroll
        for (int cch = 0; cch < 4; ++cch) {
          int kk = 4 * cch + 2 * hsel;
          // A 16x4 layout: lane -> M = mrow; VGPR0/1 -> K = kk, kk+1
          v2f aR; aR[0] = Ls_r[la + kk]; aR[1] = Ls_r[la + kk + 1];
          v2f aI; aI[0] = Ls_i[la + kk]; aI[1] = Ls_i[la + kk + 1];
          // B 4x16 layout: lane -> N = mrow; VGPR0/1 -> K = kk, kk+1
          v2f bR; bR[0] = Ut_r[lb + kk]; bR[1] = Ut_r[lb + kk + 1];
          v2f bI; bI[0] = Ut_i[lb + kk]; bI[1] = Ut_i[lb + kk + 1];
          arr = __builtin_amdgcn_wmma_f32_16x16x4_f32(false, aR, false, bR, (short)0, arr, false, false);
          aii = __builtin_amdgcn_wmma_f32_16x16x4_f32(false, aI, false, bI, (short)0, aii, false, false);
          ari = __builtin_amdgcn_wmma_f32_16x16x4_f32(false, aR, false, bI, (short)0, ari, false, false);
          air = __builtin_amdgcn_wmma_f32_16x16x4_f32(false, aI, false, bR, (short)0, air, false, false);
        }
        // C/D layout: VGPR r -> M = r + 8*hsel, N = mrow
#pragma unroll
        for (int r = 0; r < 8; ++r) {
          int row = r0 + r + 8 * hsel;
          int col = c0 + mrow;
          Mre[row * LD + col] -= arr[r] - aii[r];
          Mim[row * LD + col] -= ari[r] + air[r];
        }
      }
      if (k0 + 16 < N2)
        __builtin_prefetch(&Mre[(k0 + 16) * LD + (k0 + 16)], 0, 1);
      __threadfence_block();
      __syncthreads();
    }

    // ---- forward substitution  L z = e  (unit lower) ----
    float accr = (tid < NH) ? 1.0f : 0.0f;
    float acci = 0.0f;
    for (int j = 0; j < N2; ++j) {
      if (tid == j) { zr_s[j] = accr; zi_s[j] = acci; }
      __syncthreads();
      if (tid > j) {
        float lr = Mre[tid * LD + j], li = Mim[tid * LD + j];
        float zr = zr_s[j], zi = zi_s[j];
        accr -= lr * zr - li * zi;
        acci -= lr * zi + li * zr;
      }
    }
    __syncthreads();

    // ---- backward substitution  U v = z ----
    float wr = accr, wi = acci;
    for (int j = N2 - 1; j >= 0; --j) {
      if (tid == j) {
        float dr = Mre[j * LD + j], di = Mim[j * LD + j];
        float inv = 1.0f / (dr * dr + di * di);
        zr_s[j] = (wr * dr + wi * di) * inv;
        zi_s[j] = (wi * dr - wr * di) * inv;
      }
      __syncthreads();
      if (tid < j) {
        float ur = Mre[tid * LD + j], ui = Mim[tid * LD + j];
        float vr = zr_s[j], vi = zi_s[j];
        wr -= ur * vr - ui * vi;
        wi -= ur * vi + ui * vr;
      }
    }
    __syncthreads();

    // ---- S = sum(eta^2 |v|^2) / n^2   (w = conj(v)) ----
    float q = eta[tid] * eta[tid];
    red[tid] = q * (zr_s[tid] * zr_s[tid] + zi_s[tid] * zi_s[tid]);
    __syncthreads();
    for (int off = 128; off > 0; off >>= 1) {
      if (tid < off) red[tid] += red[tid + off];
      __syncthreads();
    }
    if (tid == 0) S[b * MM + m] = red[0] * (1.0f / (float)(NH * NH));
    __syncthreads();
  }
}

// ---------------------------------------------------------------------------
// Launch. d_out = jac (16*256*256 f32) ++ S (16*32 f32).
// Workspace: [0,16KB) y/a buffer; [64KB, 64KB+128*512KB) LU matrix slots.
// ---------------------------------------------------------------------------
extern "C" void kernel_launch(void* const* d_in, const int* in_sizes, int n_in,
                              void* d_out, int out_size, void* d_ws, size_t ws_size,
                              hipStream_t stream) {
  (void)in_sizes; (void)n_in; (void)out_size; (void)ws_size;
  const float* x        = (const float*)d_in[0];
  const float* omega    = (const float*)d_in[1];
  const float* Wzx      = (const float*)d_in[2];
  const float* log_Way  = (const float*)d_in[3];
  const float* b0       = (const float*)d_in[4];
  const float* sigma    = (const float*)d_in[5];
  const float* log_tauy = (const float*)d_in[6];
  const float* log_taua = (const float*)d_in[7];
  const float* eta      = (const float*)d_in[8];

  float* jac_out = (float*)d_out;
  float* S_out   = jac_out + (size_t)BB * MAT_ELEMS;

  float* ya   = (float*)d_ws;                          // 16*256 floats
  float* mats = (float*)((char*)d_ws + (1 << 16));     // 128 slots * 512KB

  organics_precompute<<<BB, NH, 0, stream>>>(x, Wzx, log_Way, b0, sigma, ya);
  organics_build_jac<<<dim3(N2, BB), N2, 0, stream>>>(ya, log_Way, log_tauy,
                                                      log_taua, jac_out);
  organics_solve<<<dim3(8, BB), 256, 0, stream>>>(jac_out, omega, eta, mats,
                                                  S_out);
}